// ConvNextV2Layer3d_57372173140628
// MI455X (gfx1250) — compile-verified
//
#include <hip/hip_runtime.h>

typedef __attribute__((ext_vector_type(2))) float v2f;
typedef __attribute__((ext_vector_type(8))) float v8f;

#define S    48
#define S2   2304
#define S3   110592
#define C    96
#define HC   384
#define NB   2
#define NPOS (NB * S3)          // 221184
#define NTILES32 (NPOS / 32)    // 6912 blocks for the GEMM kernels
#define EPS_LN  1e-6f
#define EPS_GRN 1e-6f

#define AS2 (C + 4)    // padded LDS stride (bank-conflict avoidance), kernel 2
#define AS4 (HC + 4)   // padded LDS stride, kernel 4

// ---------------------------------------------------------------------------
// Kernel 0: zero the per-(batch,channel) sum-of-squares accumulator
// ---------------------------------------------------------------------------
__global__ void zero_kernel(float* __restrict__ p, int n) {
    int i = blockIdx.x * blockDim.x + threadIdx.x;
    if (i < n) p[i] = 0.0f;
}

// ---------------------------------------------------------------------------
// Kernel 1: depthwise 7x7x7 conv, padding 3, + bias.
// Input x: NCDHW. Output y: channels-last (pos, c) so GEMM A-tiles coalesce.
// One block = one (b, c, 8x8x8 spatial tile); 14^3 haloed tile in LDS.
// ---------------------------------------------------------------------------
__global__ __launch_bounds__(256) void dwconv_kernel(
    const float* __restrict__ x, const float* __restrict__ dw_w,
    const float* __restrict__ dw_b, float* __restrict__ y)
{
    __shared__ float tile[14 * 14 * 14];
    __shared__ float wgt[343];

    int bid = blockIdx.x;
    int tx = bid % 6;
    int ty = (bid / 6) % 6;
    int tz = (bid / 36) % 6;
    int c  = (bid / 216) % C;
    int b  = bid / (216 * C);
    int tid = threadIdx.x;

    for (int i = tid; i < 343; i += 256) wgt[i] = dw_w[c * 343 + i];

    const float* xc = x + ((size_t)(b * C + c)) * S3;
    for (int i = tid; i < 14 * 14 * 14; i += 256) {
        int iz = i / 196, iy = (i / 14) % 14, ix = i % 14;
        int gz = tz * 8 - 3 + iz, gy = ty * 8 - 3 + iy, gx = tx * 8 - 3 + ix;
        float v = 0.0f;
        if ((unsigned)gz < (unsigned)S && (unsigned)gy < (unsigned)S &&
            (unsigned)gx < (unsigned)S)
            v = xc[gz * S2 + gy * S + gx];
        tile[i] = v;
    }
    __syncthreads();

    float bias = dw_b[c];
    for (int o = tid; o < 512; o += 256) {
        int oz = o >> 6, oy = (o >> 3) & 7, ox = o & 7;
        float acc = bias;
        for (int dz = 0; dz < 7; ++dz)
            for (int dy = 0; dy < 7; ++dy) {
                const float* tr = &tile[(oz + dz) * 196 + (oy + dy) * 14 + ox];
                const float* wr = &wgt[dz * 49 + dy * 7];
                #pragma unroll
                for (int dx = 0; dx < 7; ++dx) acc += tr[dx] * wr[dx];
            }
        int gz = tz * 8 + oz, gy = ty * 8 + oy, gx = tx * 8 + ox;
        size_t pos = (size_t)b * S3 + (size_t)(gz * S2 + gy * S + gx);
        y[pos * C + c] = acc;
    }
}

// ---------------------------------------------------------------------------
// Kernel 2 epilogue helper: bias + exact GELU + store h + LDS sumsq partial.
// D-matrix layout: VGPR v holds row M = v + 8*(lane>=16), col N = lane%16.
// ---------------------------------------------------------------------------
__device__ __forceinline__ void epilogue_tile(v8f acc, int col, size_t pbase,
                                              int kh, const float* __restrict__ b1,
                                              float* __restrict__ h,
                                              float* ssq)
{
    float bb = b1[col];
    float p = 0.0f;
    #pragma unroll
    for (int vv = 0; vv < 8; ++vv) {
        int mrow = vv + 8 * kh;
        float val = acc[vv] + bb;
        val = 0.5f * val * (1.0f + erff(val * 0.70710678118654752f)); // exact GELU
        h[(pbase + (size_t)mrow) * HC + col] = val;
        p += val * val;
    }
    atomicAdd(&ssq[col], p);
}

// ---------------------------------------------------------------------------
// Kernel 2: LayerNorm(C) + expand GEMM (32 pos x 96) @ (96 x 384) via
// v_wmma_f32_16x16x4_f32, + b1, exact GELU, store h, accumulate sum(h^2).
// 256 threads = 8 waves. Wave w owns N-tiles {3w, 3w+1, 3w+2} across BOTH
// M-tiles: each B fragment (held in regs) feeds 2 WMMAs -> half the weight
// traffic per position vs a 16-row tile.
// ---------------------------------------------------------------------------
__global__ __launch_bounds__(256) void ln_expand_kernel(
    const float* __restrict__ y, const float* __restrict__ ln_w,
    const float* __restrict__ ln_b, const float* __restrict__ w1,
    const float* __restrict__ b1, float* __restrict__ h,
    float* __restrict__ sumsq)
{
    __shared__ float A[32 * AS2];     // LN'd A tile, padded stride
    __shared__ float mu_s[32], rs_s[32];
    __shared__ float ssq[HC];

    int tid  = threadIdx.x;
    int lane = tid & 31;
    int wave = tid >> 5;
    size_t base = (size_t)blockIdx.x * 32;
    int b = (int)(base / S3);

    for (int i = tid; i < 32 * C; i += 256) {
        int m = i / C, cc = i % C;
        A[m * AS2 + cc] = y[base * C + (size_t)i];
    }
    for (int i = tid; i < HC; i += 256) ssq[i] = 0.0f;
    __syncthreads();

    if (tid < 32) {
        float s = 0.0f, s2 = 0.0f;
        for (int cc = 0; cc < C; ++cc) {
            float v = A[tid * AS2 + cc];
            s += v; s2 += v * v;
        }
        float mu  = s * (1.0f / C);
        float var = s2 * (1.0f / C) - mu * mu;
        mu_s[tid] = mu;
        rs_s[tid] = rsqrtf(var + EPS_LN);
    }
    __syncthreads();
    for (int i = tid; i < 32 * C; i += 256) {
        int m = i / C, cc = i % C;
        A[m * AS2 + cc] = (A[m * AS2 + cc] - mu_s[m]) * rs_s[m] * ln_w[cc] + ln_b[cc];
    }
    __syncthreads();

    int m  = lane & 15;   // A/B/N row-in-tile for this lane
    int kh = lane >> 4;   // K half: lanes 16-31 carry K = kbase+2, kbase+3
    int nt0 = wave * 3;

    // acc[mt][nt]: 6 accumulators
    v8f a00 = {}, a01 = {}, a02 = {};
    v8f a10 = {}, a11 = {}, a12 = {};

    __builtin_prefetch(w1 + (size_t)nt0 * 16 + m, 0, 0); // global_prefetch_b8

    for (int k = 0; k < C / 4; ++k) {
        int kr = k * 4 + kh * 2;
        v2f f0, f1;                               // A frags, M-tiles 0 and 1
        f0.x = A[m * AS2 + kr];        f0.y = A[m * AS2 + kr + 1];
        f1.x = A[(16 + m) * AS2 + kr]; f1.y = A[(16 + m) * AS2 + kr + 1];

        const float* wr = w1 + (size_t)kr * HC + m;  // rows kr, kr+1; col base m
        v2f bf0, bf1, bf2;
        bf0.x = wr[nt0 * 16];        bf0.y = wr[HC + nt0 * 16];
        bf1.x = wr[nt0 * 16 + 16];   bf1.y = wr[HC + nt0 * 16 + 16];
        bf2.x = wr[nt0 * 16 + 32];   bf2.y = wr[HC + nt0 * 16 + 32];

        a00 = __builtin_amdgcn_wmma_f32_16x16x4_f32(false, f0, false, bf0,
                                                    (short)0, a00, false, false);
        a10 = __builtin_amdgcn_wmma_f32_16x16x4_f32(false, f1, false, bf0,
                                                    (short)0, a10, false, false);
        a01 = __builtin_amdgcn_wmma_f32_16x16x4_f32(false, f0, false, bf1,
                                                    (short)0, a01, false, false);
        a11 = __builtin_amdgcn_wmma_f32_16x16x4_f32(false, f1, false, bf1,
                                                    (short)0, a11, false, false);
        a02 = __builtin_amdgcn_wmma_f32_16x16x4_f32(false, f0, false, bf2,
                                                    (short)0, a02, false, false);
        a12 = __builtin_amdgcn_wmma_f32_16x16x4_f32(false, f1, false, bf2,
                                                    (short)0, a12, false, false);
    }

    epilogue_tile(a00, nt0 * 16 + m,      base,      kh, b1, h, ssq);
    epilogue_tile(a01, nt0 * 16 + 16 + m, base,      kh, b1, h, ssq);
    epilogue_tile(a02, nt0 * 16 + 32 + m, base,      kh, b1, h, ssq);
    epilogue_tile(a10, nt0 * 16 + m,      base + 16, kh, b1, h, ssq);
    epilogue_tile(a11, nt0 * 16 + 16 + m, base + 16, kh, b1, h, ssq);
    epilogue_tile(a12, nt0 * 16 + 32 + m, base + 16, kh, b1, h, ssq);

    __syncthreads();
    for (int i = tid; i < HC; i += 256)
        atomicAdd(&sumsq[b * HC + i], ssq[i]);
}

// ---------------------------------------------------------------------------
// Kernel 3: GRN scales. g = sqrt(sumsq); n = g / (mean_e(g) + eps).
// One block per batch, HC=384 threads.
// ---------------------------------------------------------------------------
__global__ __launch_bounds__(HC) void grn_scale_kernel(
    const float* __restrict__ sumsq, float* __restrict__ nbuf)
{
    __shared__ float red[HC];
    int b = blockIdx.x, e = threadIdx.x;
    float g = sqrtf(sumsq[b * HC + e]);
    red[e] = g;
    __syncthreads();
    if (e < 128) red[e] += red[e + 256];   // 384 -> 256
    __syncthreads();
    for (int st = 128; st > 0; st >>= 1) {
        if (e < st) red[e] += red[e + st];
        __syncthreads();
    }
    float mean = red[0] * (1.0f / HC);
    nbuf[b * HC + e] = g / (mean + EPS_GRN);
}

// ---------------------------------------------------------------------------
// Kernel 4: GRN-apply + contract GEMM (32 pos x 384) @ (384 x 96) via
// v_wmma_f32_16x16x4_f32, + b2, + residual x, transposed store to NCDHW.
// 256 threads; waves 0..5 each own one 16-column tile across BOTH M-tiles,
// so each w2 fragment in registers feeds 2 WMMAs.
// ---------------------------------------------------------------------------
__global__ __launch_bounds__(256) void grn_contract_kernel(
    const float* __restrict__ h, const float* __restrict__ nbuf,
    const float* __restrict__ grn_w, const float* __restrict__ grn_b,
    const float* __restrict__ w2, const float* __restrict__ b2,
    const float* __restrict__ x, float* __restrict__ out)
{
    __shared__ float A[32 * AS4];   // GRN-applied activations, padded stride

    int tid  = threadIdx.x;
    int lane = tid & 31;
    int wave = tid >> 5;
    size_t base = (size_t)blockIdx.x * 32;
    int b = (int)(base / S3);

    for (int i = tid; i < 32 * HC; i += 256) {
        int m = i / HC, e = i % HC;
        float v  = h[base * HC + (size_t)i];
        float nv = nbuf[b * HC + e];
        A[m * AS4 + e] = grn_w[e] * (v * nv) + grn_b[e] + v;
    }
    __syncthreads();

    if (wave < 6) {
        int m  = lane & 15;
        int kh = lane >> 4;
        int col = wave * 16 + m;
        const float* wcol = w2 + col;
        __builtin_prefetch(wcol, 0, 0);

        v8f acc0 = {}, acc1 = {};
        for (int k = 0; k < HC / 4; ++k) {
            int kr = k * 4 + kh * 2;
            v2f f0, f1;
            f0.x = A[m * AS4 + kr];        f0.y = A[m * AS4 + kr + 1];
            f1.x = A[(16 + m) * AS4 + kr]; f1.y = A[(16 + m) * AS4 + kr + 1];
            v2f bf;
            bf.x = wcol[(size_t)kr * C];
            bf.y = wcol[(size_t)(kr + 1) * C];
            acc0 = __builtin_amdgcn_wmma_f32_16x16x4_f32(false, f0, false, bf,
                                                         (short)0, acc0, false, false);
            acc1 = __builtin_amdgcn_wmma_f32_16x16x4_f32(false, f1, false, bf,
                                                         (short)0, acc1, false, false);
        }

        float bb = b2[col];
        #pragma unroll
        for (int mt = 0; mt < 2; ++mt) {
            v8f acc = (mt == 0) ? acc0 : acc1;
            #pragma unroll
            for (int vv = 0; vv < 8; ++vv) {
                int mrow = vv + 8 * kh;
                size_t pos = base + (size_t)(mt * 16 + mrow);
                size_t s   = pos - (size_t)b * S3;
                size_t oi  = ((size_t)b * C + (size_t)col) * S3 + s;
                out[oi] = acc[vv] + bb + x[oi];
            }
        }
    }
}

// ---------------------------------------------------------------------------
// Launch
// ---------------------------------------------------------------------------
extern "C" void kernel_launch(void* const* d_in, const int* in_sizes, int n_in,
                              void* d_out, int out_size, void* d_ws, size_t ws_size,
                              hipStream_t stream)
{
    (void)in_sizes; (void)n_in; (void)out_size; (void)ws_size;

    const float* x     = (const float*)d_in[0];
    const float* dw_w  = (const float*)d_in[1];
    const float* dw_b  = (const float*)d_in[2];
    const float* ln_w  = (const float*)d_in[3];
    const float* ln_b  = (const float*)d_in[4];
    const float* w1    = (const float*)d_in[5];
    const float* b1    = (const float*)d_in[6];
    const float* grn_w = (const float*)d_in[7];
    const float* grn_b = (const float*)d_in[8];
    const float* w2    = (const float*)d_in[9];
    const float* b2    = (const float*)d_in[10];
    float* out = (float*)d_out;

    char* ws = (char*)d_ws;
    float* y    = (float*)ws;                                        // NPOS*C
    float* h    = (float*)(ws + (size_t)NPOS * C * sizeof(float));   // NPOS*HC
    float* ssq  = (float*)(ws + (size_t)NPOS * C * sizeof(float)
                              + (size_t)NPOS * HC * sizeof(float));  // NB*HC
    float* nbuf = ssq + NB * HC;                                     // NB*HC

    zero_kernel<<<(NB * HC + 255) / 256, 256, 0, stream>>>(ssq, NB * HC);
    dwconv_kernel<<<NB * C * 6 * 6 * 6, 256, 0, stream>>>(x, dw_w, dw_b, y);
    ln_expand_kernel<<<NTILES32, 256, 0, stream>>>(y, ln_w, ln_b, w1, b1, h, ssq);
    grn_scale_kernel<<<NB, HC, 0, stream>>>(ssq, nbuf);
    grn_contract_kernel<<<NTILES32, 256, 0, stream>>>(h, nbuf, grn_w, grn_b,
                                                      w2, b2, x, out);
}